// Embedder_28114855919902
// MI455X (gfx1250) — compile-verified
//
#include <hip/hip_runtime.h>

// ---------------------------------------------------------------------------
// Instant-NGP hash-grid encoder, MI455X (gfx1250) version.
//
// Per (point, level):  val = sum_c w[c] * s[c],  s[c] = horizontal sum of the
// corner's 16-float feature row.  A wave owns a 16-point tile; the 8-corner
// dot for all 16 points is two V_WMMA_F32_16X16X4_F32 ops (K=4 each), whose
// A (16x4 s-values) and B (4x16 weights) operands share an identical
// lane -> (point, K-pair) mapping, so each lane just computes s and w for its
// own two corners per WMMA -- zero marshalling.  Result = accumulator
// diagonal (exact f32).
//
// Tables (~97 MB) are L2-resident on MI455X (192 MB L2); the 1.07 GB of
// gathers are L2 traffic, HBM sees only ~21 MB.  VALU and L2 are roughly
// balanced, so this revision removes the 64-bit per-lane address math:
// all gather offsets are 32-bit byte offsets off a scalar (SGPR) table base,
// letting each 64-B corner row be fetched as 4 saddr-form global_load_b128
// with immediate offsets after a single v_lshlrev_b32.
// ---------------------------------------------------------------------------

typedef float v2f __attribute__((ext_vector_type(2)));
typedef float v8f __attribute__((ext_vector_type(8)));

#define NPTS 131072
#define TSZ  262147u
#define START_HASH 11

// Horizontal sum of a 16-float feature row at 64-byte-aligned byte offset
// `off` from `base` (32-bit offset -> SGPR-base + VGPR-offset addressing).
__device__ __forceinline__ float sum16(const float* __restrict__ base,
                                       unsigned off) {
  const float4* q = (const float4*)((const char*)base + off);
  float4 a = q[0], b = q[1], c = q[2], d = q[3];
  float s0 = (a.x + a.y) + (a.z + a.w);
  float s1 = (b.x + b.y) + (b.z + b.w);
  float s2 = (c.x + c.y) + (c.z + c.w);
  float s3 = (d.x + d.y) + (d.z + d.w);
  return (s0 + s1) + (s2 + s3);
}

__global__ __launch_bounds__(256) void ngp_wmma_kernel(
    const float* __restrict__ xyz,
    const float* __restrict__ dense,
    const float* __restrict__ hasht,
    float* __restrict__ out)
{
  // floor(2 * 1.38^i)
  constexpr int   EN[16]   = {2,2,3,5,7,10,13,19,26,36,50,69,95,131,181,250};
  // exclusive cumsum of EN^3 for the dense levels (levels 0..10)
  constexpr int   LOFF[11] = {0,8,16,43,168,511,1511,3708,10567,28143,74799};
  // scale = (n-1); reference divides by float32(1/(n-1)); multiply is <=1 ulp off
  constexpr float SC[16] = {1.f,1.f,2.f,4.f,6.f,9.f,12.f,18.f,
                            25.f,35.f,49.f,68.f,94.f,130.f,180.f,249.f};

  const int lane = threadIdx.x & 31;
  const int wave = threadIdx.x >> 5;          // 0..7
  const int p    = lane & 15;                 // point within 16-pt tile
  const int hi   = lane >> 4;                 // K-pair half (0: K=0,1 ; 1: K=2,3)
  const int ptBase = (blockIdx.x * 8 + wave) * 16;
  const int pt = ptBase + p;

  const unsigned xoff = (unsigned)(3 * pt) * 4u;            // byte offset into xyz
  const float x = *(const float*)((const char*)xyz + xoff);
  const float y = *(const float*)((const char*)xyz + xoff + 4);
  const float z = *(const float*)((const char*)xyz + xoff + 8);

  const unsigned obase = (unsigned)(19 * pt) * 4u;          // byte offset into out
  // xyz passthrough (cols 0..2); lanes 16..31 duplicate points, skip them.
  if (hi == 0) {
    *(float*)((char*)out + obase + 0) = x;
    *(float*)((char*)out + obase + 4) = y;
    *(float*)((char*)out + obase + 8) = z;
  }

  #pragma unroll
  for (int l = 0; l < 16; ++l) {
    const int n = EN[l];
    const float fx = x * SC[l];
    const float fy = y * SC[l];
    const float fz = z * SC[l];

    int ix = (int)fx; ix = ix > n-1 ? n-1 : ix;
    int iy = (int)fy; iy = iy > n-1 ? n-1 : iy;
    int iz = (int)fz; iz = iz > n-1 ? n-1 : iz;

    const float ox = fx - (float)ix;
    const float oy = fy - (float)iy;
    const float oz = fz - (float)iz;

    const int ix1 = (ix+1 > n-1) ? n-1 : ix+1;
    const int iy1 = (iy+1 > n-1) ? n-1 : iy+1;
    const int iz1 = (iz+1 > n-1) ? n-1 : iz+1;

    // This lane's 4 corners share y-bit == hi (corner bits: x=c>>2, y=(c>>1)&1,
    // z=c&1; corners {2hi, 2hi+1} feed WMMA #1, {2hi+4, 2hi+5} feed WMMA #2).
    const int   jy = hi ? iy1 : iy;
    const float wy = hi ? oy  : (1.0f - oy);
    const float wx0 = 1.0f - ox, wx1 = ox;
    const float wz0 = 1.0f - oz, wz1 = oz;

    // Corner-row horizontal sums s(xbit, zbit); all offsets 32-bit bytes.
    float s00, s01, s10, s11;
    if (l < START_HASH) {
      const unsigned n2   = (unsigned)(n * n);
      const unsigned base = (unsigned)(LOFF[l] + jy * n);
      const unsigned tx0  = (unsigned)ix  * n2;
      const unsigned tx1  = (unsigned)ix1 * n2;
      s00 = sum16(dense, (base + tx0 + (unsigned)iz ) << 6);
      s01 = sum16(dense, (base + tx0 + (unsigned)iz1) << 6);
      s10 = sum16(dense, (base + tx1 + (unsigned)iz ) << 6);
      s11 = sum16(dense, (base + tx1 + (unsigned)iz1) << 6);
    } else {
      // reference: (x*1 ^ y*2654435761 ^ z*805459861) % 262147 in int64 (all >=0)
      const unsigned long long hx0 = (unsigned long long)(unsigned)ix;
      const unsigned long long hx1 = (unsigned long long)(unsigned)ix1;
      const unsigned long long hy  = (unsigned long long)(unsigned)jy  * 2654435761ull;
      const unsigned long long hz0 = (unsigned long long)(unsigned)iz  * 805459861ull;
      const unsigned long long hz1 = (unsigned long long)(unsigned)iz1 * 805459861ull;
      const float* tab = hasht + (size_t)(l - START_HASH) * ((size_t)TSZ * 16u);
      const unsigned h00 = (unsigned)((hx0 ^ hy ^ hz0) % TSZ);
      const unsigned h01 = (unsigned)((hx0 ^ hy ^ hz1) % TSZ);
      const unsigned h10 = (unsigned)((hx1 ^ hy ^ hz0) % TSZ);
      const unsigned h11 = (unsigned)((hx1 ^ hy ^ hz1) % TSZ);
      s00 = sum16(tab, h00 << 6);
      s01 = sum16(tab, h01 << 6);
      s10 = sum16(tab, h10 << 6);
      s11 = sum16(tab, h11 << 6);
    }

    const float w00 = wx0 * wy * wz0;
    const float w01 = wx0 * wy * wz1;
    const float w10 = wx1 * wy * wz0;
    const float w11 = wx1 * wy * wz1;

    // A (16x4 f32): lane L holds s[p, 2hi], s[p, 2hi+1] in its 2 VGPRs.
    // B (4x16 f32): lane L holds w[p, 2hi], w[p, 2hi+1] -- identical mapping.
    v8f acc = {};
    v2f A0; A0.x = s00; A0.y = s01;
    v2f B0; B0.x = w00; B0.y = w01;
    acc = __builtin_amdgcn_wmma_f32_16x16x4_f32(false, A0, false, B0,
                                                (short)0, acc, false, false);
    v2f A1; A1.x = s10; A1.y = s11;   // corners 4..7 -> second K=4 pass
    v2f B1; B1.x = w10; B1.y = w11;
    acc = __builtin_amdgcn_wmma_f32_16x16x4_f32(false, A1, false, B1,
                                                (short)0, acc, false, false);

    // Diagonal D[p,p]: lanes 0..7 hold it in VGPR=lane, lanes 24..31 in
    // VGPR=lane-24 (C/D layout: VGPR r, lanes 0-15 -> M=r; lanes 16-31 -> M=8+r).
    const int dp = lane & 7;
    float v = acc[0];
    if (dp == 1) v = acc[1];
    if (dp == 2) v = acc[2];
    if (dp == 3) v = acc[3];
    if (dp == 4) v = acc[4];
    if (dp == 5) v = acc[5];
    if (dp == 6) v = acc[6];
    if (dp == 7) v = acc[7];

    const bool active = (lane < 8) || (lane >= 24);
    if (active) {
      const int prow = (lane < 8) ? lane : (lane - 16);
      const unsigned oo = (unsigned)(19 * (ptBase + prow) + 3 + l) * 4u;
      *(float*)((char*)out + oo) = v;
    }
  }
}

extern "C" void kernel_launch(void* const* d_in, const int* in_sizes, int n_in,
                              void* d_out, int out_size, void* d_ws, size_t ws_size,
                              hipStream_t stream) {
  (void)in_sizes; (void)n_in; (void)out_size; (void)d_ws; (void)ws_size;
  const float* xyz   = (const float*)d_in[0];
  const float* dense = (const float*)d_in[1];
  const float* hasht = (const float*)d_in[2];
  float* out = (float*)d_out;

  // 256 threads = 8 waves/block, 16 points per wave -> 128 points/block.
  dim3 grid(NPTS / 128);
  ngp_wmma_kernel<<<grid, dim3(256), 0, stream>>>(xyz, dense, hasht, out);
}